// SplitTableBatchedEmbeddingBagsCodegen_65369402245265
// MI455X (gfx1250) — compile-verified
//
#include <hip/hip_runtime.h>
#include <hip/hip_bf16.h>
#include <stdint.h>

// Problem constants (from the reference): T=4, E=250000, D=128, B=4096, L=50
#define T_TABLES 4
#define E_ROWS   250000
#define D_DIM    128
#define B_BATCH  4096
#define L_BAG    50

#define WAVES_PER_BLOCK 8
#define ROWS_PER_TDM    8                          // 32-bit gather indices -> max 8 rows per D#
#define CHUNKS          ((L_BAG + ROWS_PER_TDM - 1) / ROWS_PER_TDM)   // 7
#define LAST_NV         (L_BAG - ROWS_PER_TDM * (CHUNKS - 1))          // 2
#define CHUNK_FLOATS    (ROWS_PER_TDM * D_DIM)     // 1024 floats = 4KB
#define CHUNK_BYTES     (CHUNK_FLOATS * 4)

typedef __attribute__((ext_vector_type(4))) unsigned int u32x4;
typedef __attribute__((ext_vector_type(8))) int          i32x8;
typedef __attribute__((ext_vector_type(4))) int          i32x4;
typedef __attribute__((ext_vector_type(4))) float        f32x4;

// Issue one TDM gather of up to 8 embedding rows (512B each) into LDS.
// All inputs are wave-uniform -> descriptor lives in SGPRs.
__device__ __forceinline__ void tdm_gather_rows(unsigned lds_addr,
                                                unsigned long long table_addr,
                                                int nvalid,
                                                const int* __restrict__ idxp,
                                                int chunk) {
    int rows[ROWS_PER_TDM];
#pragma unroll
    for (int j = 0; j < ROWS_PER_TDM; ++j) {
        const int l = chunk * ROWS_PER_TDM + j;
        rows[j] = (l < L_BAG) ? idxp[l] : 0;       // uniform scalar loads (KMcnt)
    }

    // ---- D# group 0 (128b): control | lds_addr | global_addr | type ----
    u32x4 g0;
    g0[0] = 0xC0000001u;                           // count=1, gather_index=32b, gather_mode=1
    g0[1] = lds_addr;                              // LDS byte address of tile start
    g0[2] = (unsigned)(table_addr & 0xFFFFFFFFull);            // global_addr[31:0]
    g0[3] = (unsigned)((table_addr >> 32) & 0x1FFFFFFull)      // global_addr[56:32]
            | 0x80000000u;                         // type=2 ("image")

    // ---- D# group 1 (256b): sizes/strides ----
    i32x8 g1;
    g1[0] = (2 << 16);                             // wg_mask=0 (not in cluster), data_size=4B
    g1[1] = (D_DIM & 0xFFFF) << 16;                // tensor_dim0[15:0] = 128 (bits 79:48)
    g1[2] = (int)(((unsigned)(E_ROWS & 0xFFFF)) << 16);  // dim0[31:16]=0 | tensor_dim1[15:0]
    g1[3] = (D_DIM << 16) | ((E_ROWS >> 16) & 0xFFFF);   // tile_dim0=128 | tensor_dim1[31:16]
    g1[4] = (nvalid & 0xFFFF);                     // tile_dim1 = #valid gather indices
    g1[5] = D_DIM;                                 // tensor_dim0_stride = 128 elements
    g1[6] = 0;                                     // stride[47:32]=0 | dim1_stride ignored
    g1[7] = 0;

    // ---- D# groups 2/3: 32-bit row indices 0..7 ----
    i32x4 g2 = { rows[0], rows[1], rows[2], rows[3] };
    i32x4 g3 = { rows[4], rows[5], rows[6], rows[7] };

    // 6-arg toolchain variant: trailing int32x8 group is unused -> zero fill.
    i32x8 g4 = { 0, 0, 0, 0, 0, 0, 0, 0 };

    __builtin_amdgcn_tensor_load_to_lds(g0, g1, g2, g3, g4, 0);
}

__global__ __launch_bounds__(WAVES_PER_BLOCK * 32)
void SplitTableBatchedEmbeddingBags_tdm_gather_kernel(
        const float* __restrict__ weights,   // [T, E, D]
        const int*   __restrict__ indices,   // [T, B, L]
        float*       __restrict__ out)       // [B, T*D]
{
    __shared__ __align__(16) float lds[WAVES_PER_BLOCK][2][CHUNK_FLOATS];

    const int lane = threadIdx.x & 31;
    const int wave = __builtin_amdgcn_readfirstlane((int)(threadIdx.x >> 5));
    const int bag  = blockIdx.x * WAVES_PER_BLOCK + wave;   // 0 .. T*B-1 (exact grid)
    const int t    = bag >> 12;                             // bag / B_BATCH
    const int b    = bag & (B_BATCH - 1);

    const int* idxp = indices + (size_t)bag * L_BAG;        // wave-uniform pointer
    const unsigned long long table_addr =
        (unsigned long long)(uintptr_t)(weights + (size_t)t * E_ROWS * D_DIM);

    // LDS byte address: low 32 bits of the flat address are the LDS offset.
    const unsigned lbase =
        (unsigned)(uintptr_t)(void*)&lds[wave][0][0];

    // Prime the pipeline: chunk 0 -> buffer 0.
    tdm_gather_rows(lbase, table_addr, ROWS_PER_TDM, idxp, 0);

    f32x4 acc = {0.f, 0.f, 0.f, 0.f};

#pragma unroll
    for (int c = 0; c < CHUNKS; ++c) {
        if (c + 1 < CHUNKS) {
            const int nv_next = (c + 1 == CHUNKS - 1) ? LAST_NV : ROWS_PER_TDM;
            tdm_gather_rows(lbase + (unsigned)(((c + 1) & 1) * CHUNK_BYTES),
                            table_addr, nv_next, idxp, c + 1);
            __builtin_amdgcn_s_wait_tensorcnt(1);   // chunk c complete (in-order TDM)
        } else {
            __builtin_amdgcn_s_wait_tensorcnt(0);   // drain
        }
        asm volatile("" ::: "memory");              // keep LDS reads after the wait

        const float* src = &lds[wave][c & 1][0];
        const int nv = (c == CHUNKS - 1) ? LAST_NV : ROWS_PER_TDM;
#pragma unroll
        for (int r = 0; r < ROWS_PER_TDM; ++r) {
            if (r < nv) {
                f32x4 v = *(const f32x4*)(src + r * D_DIM + lane * 4);
                acc += v;                            // ds_load_b128 + v_add_f32 x4
            }
        }
    }

    // out[b][t*D + d], d = lane*4 .. lane*4+3
    float* op = out + (size_t)b * (T_TABLES * D_DIM) + t * D_DIM + lane * 4;
    *(f32x4*)op = acc;                               // global_store_b128
}

extern "C" void kernel_launch(void* const* d_in, const int* in_sizes, int n_in,
                              void* d_out, int out_size, void* d_ws, size_t ws_size,
                              hipStream_t stream) {
    const float* weights = (const float*)d_in[0];   // [T,E,D] fp32
    const int*   indices = (const int*)d_in[1];     // [T,B,L] int32
    float*       out     = (float*)d_out;           // [B, T*D] fp32

    const int total_bags = T_TABLES * B_BATCH;      // 16384
    dim3 grid(total_bags / WAVES_PER_BLOCK);        // 2048 blocks
    dim3 block(WAVES_PER_BLOCK * 32);               // 256 threads = 8 waves

    hipLaunchKernelGGL(SplitTableBatchedEmbeddingBags_tdm_gather_kernel,
                       grid, block, 0, stream, weights, indices, out);
}